// GatedGCN_26680336842924
// MI455X (gfx1250) — compile-verified
//
#include <hip/hip_runtime.h>
#include <hip/hip_bf16.h>

typedef __attribute__((ext_vector_type(2))) float v2f;
typedef __attribute__((ext_vector_type(8))) float v8f;

#define DD 96              // feature dim == units
#define KSTEPS (DD / 4)    // 24 wmma k-steps
#define CTILES (DD / 16)   // 6 column tiles of 16
#define LDS_PAIR_STRIDE (2 * DD)   // pair-interleaved weight rows

// ---------------------------------------------------------------------------
// LDS weight staging, pair-interleaved so that a B fragment (W[k][n],
// W[k+1][n]) is one aligned ds_load_b64:
//   lds[(k>>1)*192 + n*2 + (k&1)] = W[k][n]
// ---------------------------------------------------------------------------
__device__ __forceinline__ void stage_weights(float* __restrict__ lds,
                                              const float* __restrict__ W) {
    for (int i = threadIdx.x; i < DD * DD; i += 128) {
        const int r = i / DD;
        const int c = i - r * DD;
        lds[(r >> 1) * LDS_PAIR_STRIDE + c * 2 + (r & 1)] = W[i];
    }
}

// ---------------------------------------------------------------------------
// Kernel 1: zero the aggregation buffer (workspace is poisoned, must re-init
// every call for determinism).
// ---------------------------------------------------------------------------
__global__ void zero_kernel(float* __restrict__ p, int n) {
    int i = blockIdx.x * blockDim.x + threadIdx.x;
    if (i < n) p[i] = 0.0f;
}

// ---------------------------------------------------------------------------
// Kernel 2: newX = X @ Wn + bn   (fp32 WMMA 16x16x4)
// Block = 128 threads = 4 waves. Each wave owns TWO 16-row tiles (32 rows),
// so every B fragment fetched from LDS feeds two WMMAs.
//
// Fragment layouts (wave32, CDNA5 ISA 7.12.2):
//   A 16x4 f32 : lane(l16)=M, a[v] holds K = 2*half + v   (half = lane>>4)
//   B 4x16 f32 : lane(l16)=N, b[v] holds K = 2*half + v
//   C 16x16 f32: lane(l16)=N, c[r] holds M = r + 8*half
// ---------------------------------------------------------------------------
__global__ __launch_bounds__(128) void gemm_bias_kernel(
    const float* __restrict__ X, const float* __restrict__ W,
    const float* __restrict__ bias, float* __restrict__ Y,
    int nPairs, int N)
{
    __shared__ float wLds[DD * DD];
    stage_weights(wLds, W);
    __syncthreads();

    const int lane = threadIdx.x & 31;
    const int wave = threadIdx.x >> 5;
    const int half = lane >> 4;
    const int l16  = lane & 15;
    const int pair = blockIdx.x * 4 + wave;
    if (pair >= nPairs) return;                 // wave-uniform, EXEC stays full
    const int rowBase = pair * 32;

    int m0 = rowBase + l16;       if (m0 >= N) m0 = N - 1;
    int m1 = rowBase + 16 + l16;  if (m1 >= N) m1 = N - 1;
    const float* aPtr0 = X + (size_t)m0 * DD + 2 * half;
    const float* aPtr1 = X + (size_t)m1 * DD + 2 * half;

    v8f acc0[CTILES] = {};
    v8f acc1[CTILES] = {};
    for (int kt = 0; kt < KSTEPS; ++kt) {
        const int k = kt * 4;
        v2f a0 = *(const v2f*)(aPtr0 + k);
        v2f a1 = *(const v2f*)(aPtr1 + k);
        const float* bRow = wLds + (size_t)(kt * 2 + half) * LDS_PAIR_STRIDE;
        #pragma unroll
        for (int ct = 0; ct < CTILES; ++ct) {
            v2f b = *(const v2f*)(bRow + (ct * 16 + l16) * 2); // one ds_load_b64
            acc0[ct] = __builtin_amdgcn_wmma_f32_16x16x4_f32(
                false, a0, false, b, (short)0, acc0[ct], false, false);
            acc1[ct] = __builtin_amdgcn_wmma_f32_16x16x4_f32(
                false, a1, false, b, (short)0, acc1[ct], false, false);
        }
    }

    #pragma unroll
    for (int ct = 0; ct < CTILES; ++ct) {
        const int n = ct * 16 + l16;
        const float bv = bias[n];
        #pragma unroll
        for (int r = 0; r < 8; ++r) {
            const int ma = rowBase + r + 8 * half;
            const int mb = rowBase + 16 + r + 8 * half;
            if (ma < N) Y[(size_t)ma * DD + n] = acc0[ct][r] + bv;
            if (mb < N) Y[(size_t)mb * DD + n] = acc1[ct][r] + bv;
        }
    }
}

// ---------------------------------------------------------------------------
// Kernel 3: SpMM scatter. agg[row[e]] += a_vals[e] * newX[col[e]]
// One thread per (edge, float4 chunk): 24 chunks of 4 floats cover U=96.
// newX (19.2 MB) is L2-resident on MI455X (192 MB L2), so the random gather
// and the atomics are L2-bandwidth bound, not HBM bound.
// ---------------------------------------------------------------------------
__global__ void scatter_kernel(
    const float* __restrict__ newX, const float* __restrict__ avals,
    const int* __restrict__ row, const int* __restrict__ col,
    float* __restrict__ agg, int E)
{
    const int t = blockIdx.x * blockDim.x + threadIdx.x;
    const int e = t / 24;
    const int c = t % 24;
    if (e >= E) return;
    const float av = avals[e];
    const int dst = row[e];
    const int src = col[e];
    const float4 v = *(const float4*)(newX + (size_t)src * DD + c * 4);
    float* p = agg + (size_t)dst * DD + c * 4;
    atomicAdd(p + 0, av * v.x);
    atomicAdd(p + 1, av * v.y);
    atomicAdd(p + 2, av * v.z);
    atomicAdd(p + 3, av * v.w);
}

// ---------------------------------------------------------------------------
// Kernel 4: fused gate GEMMs + blend.
//   G   = X @ Wgi + agg @ Wgn + bgi + bgn
//   out = agg * sigmoid(G) + X * (1 - sigmoid(G))
// Two WMMA K-loops share one 36 KB LDS weight buffer (barrier-reload between
// phases so all waves hit every __syncthreads()). Two row tiles per wave.
// ---------------------------------------------------------------------------
__global__ __launch_bounds__(128) void gate_kernel(
    const float* __restrict__ X, const float* __restrict__ agg,
    const float* __restrict__ Wgi, const float* __restrict__ Wgn,
    const float* __restrict__ bgi, const float* __restrict__ bgn,
    float* __restrict__ out, int nPairs, int N)
{
    __shared__ float wLds[DD * DD];

    const int lane = threadIdx.x & 31;
    const int wave = threadIdx.x >> 5;
    const int half = lane >> 4;
    const int l16  = lane & 15;
    const int pair = blockIdx.x * 4 + wave;
    const bool active = (pair < nPairs);        // wave-uniform
    const int rowBase = pair * 32;
    int m0 = rowBase + l16;       if (m0 >= N) m0 = (N > 0) ? N - 1 : 0;
    int m1 = rowBase + 16 + l16;  if (m1 >= N) m1 = (N > 0) ? N - 1 : 0;

    v8f acc0[CTILES] = {};
    v8f acc1[CTILES] = {};

    // ---- phase 1: acc += X @ Wgi ----
    stage_weights(wLds, Wgi);
    __syncthreads();
    if (active) {
        const float* aPtr0 = X + (size_t)m0 * DD + 2 * half;
        const float* aPtr1 = X + (size_t)m1 * DD + 2 * half;
        for (int kt = 0; kt < KSTEPS; ++kt) {
            const int k = kt * 4;
            v2f a0 = *(const v2f*)(aPtr0 + k);
            v2f a1 = *(const v2f*)(aPtr1 + k);
            const float* bRow = wLds + (size_t)(kt * 2 + half) * LDS_PAIR_STRIDE;
            #pragma unroll
            for (int ct = 0; ct < CTILES; ++ct) {
                v2f b = *(const v2f*)(bRow + (ct * 16 + l16) * 2);
                acc0[ct] = __builtin_amdgcn_wmma_f32_16x16x4_f32(
                    false, a0, false, b, (short)0, acc0[ct], false, false);
                acc1[ct] = __builtin_amdgcn_wmma_f32_16x16x4_f32(
                    false, a1, false, b, (short)0, acc1[ct], false, false);
            }
        }
    }
    __syncthreads();

    // ---- phase 2: acc += agg @ Wgn ----
    stage_weights(wLds, Wgn);
    __syncthreads();
    if (active) {
        const float* aPtr0 = agg + (size_t)m0 * DD + 2 * half;
        const float* aPtr1 = agg + (size_t)m1 * DD + 2 * half;
        for (int kt = 0; kt < KSTEPS; ++kt) {
            const int k = kt * 4;
            v2f a0 = *(const v2f*)(aPtr0 + k);
            v2f a1 = *(const v2f*)(aPtr1 + k);
            const float* bRow = wLds + (size_t)(kt * 2 + half) * LDS_PAIR_STRIDE;
            #pragma unroll
            for (int ct = 0; ct < CTILES; ++ct) {
                v2f b = *(const v2f*)(bRow + (ct * 16 + l16) * 2);
                acc0[ct] = __builtin_amdgcn_wmma_f32_16x16x4_f32(
                    false, a0, false, b, (short)0, acc0[ct], false, false);
                acc1[ct] = __builtin_amdgcn_wmma_f32_16x16x4_f32(
                    false, a1, false, b, (short)0, acc1[ct], false, false);
            }
        }

        // ---- epilogue: sigmoid gate + residual blend ----
        #pragma unroll
        for (int ct = 0; ct < CTILES; ++ct) {
            const int n = ct * 16 + l16;
            const float bsum = bgi[n] + bgn[n];
            #pragma unroll
            for (int r = 0; r < 8; ++r) {
                #pragma unroll
                for (int t2 = 0; t2 < 2; ++t2) {
                    const int m = rowBase + t2 * 16 + r + 8 * half;
                    if (m < N) {
                        const float accv = (t2 == 0) ? acc0[ct][r] : acc1[ct][r];
                        float g = accv + bsum;
                        g = 1.0f / (1.0f + __expf(-g));
                        const float av = agg[(size_t)m * DD + n];
                        const float xv = X[(size_t)m * DD + n];
                        out[(size_t)m * DD + n] = av * g + xv * (1.0f - g);
                    }
                }
            }
        }
    }
}

// ---------------------------------------------------------------------------
// Launch: inputs (setup_inputs order):
//   0:X [N*96] 1:a_vals [E] 2:Wn [96*96] 3:bn [96] 4:Wgi 5:bgi 6:Wgn 7:bgn
//   8:row [E] 9:col [E]
// Workspace: newX (N*96 f32) | agg (N*96 f32)
// ---------------------------------------------------------------------------
extern "C" void kernel_launch(void* const* d_in, const int* in_sizes, int n_in,
                              void* d_out, int out_size, void* d_ws, size_t ws_size,
                              hipStream_t stream) {
    const float* X      = (const float*)d_in[0];
    const float* a_vals = (const float*)d_in[1];
    const float* Wn     = (const float*)d_in[2];
    const float* bn     = (const float*)d_in[3];
    const float* Wgi    = (const float*)d_in[4];
    const float* bgi    = (const float*)d_in[5];
    const float* Wgn    = (const float*)d_in[6];
    const float* bgn    = (const float*)d_in[7];
    const int*   row    = (const int*)d_in[8];
    const int*   col    = (const int*)d_in[9];
    float* out = (float*)d_out;

    const int N = in_sizes[0] / DD;
    const int E = in_sizes[1];

    float* newX = (float*)d_ws;
    float* agg  = (float*)d_ws + (size_t)N * DD;

    // 1) zero agg
    {
        const int total = N * DD;
        zero_kernel<<<(total + 255) / 256, 256, 0, stream>>>(agg, total);
    }

    const int nPairs = (N + 31) / 32;
    const int gemmBlocks = (nPairs + 3) / 4;

    // 2) newX = X @ Wn + bn
    gemm_bias_kernel<<<gemmBlocks, 128, 0, stream>>>(X, Wn, bn, newX, nPairs, N);

    // 3) agg = A @ newX (gather + atomic scatter)
    {
        const long long threads = (long long)E * 24;
        const int blocks = (int)((threads + 255) / 256);
        scatter_kernel<<<blocks, 256, 0, stream>>>(newX, a_vals, row, col, agg, E);
    }

    // 4) gate + blend
    gate_kernel<<<gemmBlocks, 128, 0, stream>>>(X, agg, Wgi, Wgn, bgi, bgn,
                                                out, nPairs, N);
}